// SpatialTransformer_range_flow_flow_list222_85555748536623
// MI455X (gfx1250) — compile-verified
//
#include <hip/hip_runtime.h>
#include <math.h>

namespace {

typedef __attribute__((ext_vector_type(4))) float v4f;  // native vec -> NT builtins OK

constexpr int Dv  = 160, Hv = 192, Wv = 160;
constexpr int HW  = Hv * Wv;        // 30720
constexpr int DHW = Dv * Hv * Wv;   // 4915200
constexpr int N4  = DHW / 4;        // 1228800 (== 4800 * 256)

__device__ __forceinline__ int iclamp(int v, int lo, int hi) {  // -> v_med3_i32
  return min(max(v, lo), hi);
}

// 3-channel trilinear sample of flow1 with zero padding. Validity folded into
// per-axis weights; addresses clamped with integer med3. Early-out before
// building weights: the unnormalized grid makes this sample OOB for
// essentially every voxel (wave-uniform skip).
__device__ __forceinline__ void sample_flow1(const float* __restrict__ f1,
                                             float ix, float iy, float iz,
                                             float& s0, float& s1, float& s2) {
  s0 = 0.f; s1 = 0.f; s2 = 0.f;
  const float fx = floorf(ix), fy = floorf(iy), fz = floorf(iz);
  const int x0 = (int)fx, y0 = (int)fy, z0 = (int)fz;   // saturating v_cvt_i32_f32
  const int x1 = x0 + 1,  y1 = y0 + 1,  z1 = z0 + 1;
  const bool vx0 = (unsigned)x0 < (unsigned)Wv, vx1 = (unsigned)x1 < (unsigned)Wv;
  const bool vy0 = (unsigned)y0 < (unsigned)Hv, vy1 = (unsigned)y1 < (unsigned)Hv;
  const bool vz0 = (unsigned)z0 < (unsigned)Dv, vz1 = (unsigned)z1 < (unsigned)Dv;
  if (!((vx0 | vx1) & (vy0 | vy1) & (vz0 | vz1))) return;

  const float tx = ix - fx, ty = iy - fy, tz = iz - fz;
  const float wx0 = vx0 ? 1.f - tx : 0.f, wx1 = vx1 ? tx : 0.f;
  const float wy0 = vy0 ? 1.f - ty : 0.f, wy1 = vy1 ? ty : 0.f;
  const float wz0 = vz0 ? 1.f - tz : 0.f, wz1 = vz1 ? tz : 0.f;

  const int cx0 = iclamp(x0, 0, Wv - 1), cx1 = iclamp(x1, 0, Wv - 1);
  const int cy0 = iclamp(y0, 0, Hv - 1), cy1 = iclamp(y1, 0, Hv - 1);
  const int cz0 = iclamp(z0, 0, Dv - 1), cz1 = iclamp(z1, 0, Dv - 1);

  const int b00 = cz0 * HW + cy0 * Wv;
  const int b01 = cz0 * HW + cy1 * Wv;
  const int b10 = cz1 * HW + cy0 * Wv;
  const int b11 = cz1 * HW + cy1 * Wv;

  const float wzy00 = wz0 * wy0, wzy01 = wz0 * wy1;
  const float wzy10 = wz1 * wy0, wzy11 = wz1 * wy1;

  // reference accumulation order: dz outer, dy, dx inner
  const int   lin[8] = { b00 + cx0, b00 + cx1, b01 + cx0, b01 + cx1,
                         b10 + cx0, b10 + cx1, b11 + cx0, b11 + cx1 };
  const float wgt[8] = { wzy00 * wx0, wzy00 * wx1, wzy01 * wx0, wzy01 * wx1,
                         wzy10 * wx0, wzy10 * wx1, wzy11 * wx0, wzy11 * wx1 };
#pragma unroll
  for (int k = 0; k < 8; ++k) {
    const int   l = lin[k];
    const float w = wgt[k];
    s0 = fmaf(w, f1[l],           s0);
    s1 = fmaf(w, f1[l + DHW],     s1);
    s2 = fmaf(w, f1[l + 2 * DHW], s2);
  }
}

// Single-channel trilinear sample of src (same tap construction).
__device__ __forceinline__ float sample_src(const float* __restrict__ src,
                                            float jx, float jy, float jz) {
  const float fx = floorf(jx), fy = floorf(jy), fz = floorf(jz);
  const int x0 = (int)fx, y0 = (int)fy, z0 = (int)fz;
  const int x1 = x0 + 1,  y1 = y0 + 1,  z1 = z0 + 1;
  const bool vx0 = (unsigned)x0 < (unsigned)Wv, vx1 = (unsigned)x1 < (unsigned)Wv;
  const bool vy0 = (unsigned)y0 < (unsigned)Hv, vy1 = (unsigned)y1 < (unsigned)Hv;
  const bool vz0 = (unsigned)z0 < (unsigned)Dv, vz1 = (unsigned)z1 < (unsigned)Dv;
  if (!((vx0 | vx1) & (vy0 | vy1) & (vz0 | vz1))) return 0.f;

  const float tx = jx - fx, ty = jy - fy, tz = jz - fz;
  const float wx0 = vx0 ? 1.f - tx : 0.f, wx1 = vx1 ? tx : 0.f;
  const float wy0 = vy0 ? 1.f - ty : 0.f, wy1 = vy1 ? ty : 0.f;
  const float wz0 = vz0 ? 1.f - tz : 0.f, wz1 = vz1 ? tz : 0.f;

  const int cx0 = iclamp(x0, 0, Wv - 1), cx1 = iclamp(x1, 0, Wv - 1);
  const int cy0 = iclamp(y0, 0, Hv - 1), cy1 = iclamp(y1, 0, Hv - 1);
  const int cz0 = iclamp(z0, 0, Dv - 1), cz1 = iclamp(z1, 0, Dv - 1);

  const int b00 = cz0 * HW + cy0 * Wv;
  const int b01 = cz0 * HW + cy1 * Wv;
  const int b10 = cz1 * HW + cy0 * Wv;
  const int b11 = cz1 * HW + cy1 * Wv;

  const float wzy00 = wz0 * wy0, wzy01 = wz0 * wy1;
  const float wzy10 = wz1 * wy0, wzy11 = wz1 * wy1;

  float acc;
  acc = (wzy00 * wx0) * src[b00 + cx0];
  acc = fmaf(wzy00 * wx1, src[b00 + cx1], acc);
  acc = fmaf(wzy01 * wx0, src[b01 + cx0], acc);
  acc = fmaf(wzy01 * wx1, src[b01 + cx1], acc);
  acc = fmaf(wzy10 * wx0, src[b10 + cx0], acc);
  acc = fmaf(wzy10 * wx1, src[b10 + cx1], acc);
  acc = fmaf(wzy11 * wx0, src[b11 + cx0], acc);
  acc = fmaf(wzy11 * wx1, src[b11 + cx1], acc);
  return acc;
}

// 4 consecutive w-voxels per thread. Streaming traffic is b128 and tagged
// non-temporal so the write-once outputs (79MB) and read-once flow2 (59MB)
// don't evict the gather-side src+flow1 (79MB) from the 192MB L2.
__global__ void __launch_bounds__(256)
warp3d_fused_v6(const float* __restrict__ src,
                const float* __restrict__ flow1,
                const v4f* __restrict__ flow2v,
                const float* __restrict__ range_flow_p,
                v4f* __restrict__ out_deform,
                v4f* __restrict__ out_flow) {
  const int idx = blockIdx.x * blockDim.x + threadIdx.x;
  if (idx >= N4) return;

  const float rf = range_flow_p[0];   // uniform -> scalar load

  // Folded coordinate transforms (exact algebra of the reference chains):
  //   sample1: ix = 80*gx+79.5, gx = wc + rf*f2w  ->  ix = (80*rf)*f2w + (80*wc+79.5)
  //   sample2: jx = (160/159)*(wc + rf*ow) - 0.5  (divisions eliminated)
  const float iAx = 80.f * rf;                 // Wv/2 * rf
  const float iAy = 96.f * rf;                 // Hv/2 * rf
  const float iAz = 80.f * rf;                 // Dv/2 * rf
  const float jSx = rf * (160.f / 159.f);      // Wv/(Wv-1) * rf
  const float jSy = rf * (192.f / 191.f);      // Hv/(Hv-1) * rf
  const float jSz = rf * (160.f / 159.f);      // Dv/(Dv-1) * rf

  const int base = idx * 4;            // Wv % 4 == 0 -> all 4 voxels share h,d
  const int w0   = base % Wv;
  const int h    = (base / Wv) % Hv;
  const int d    = base / HW;
  const float hc = (float)h, dc = (float)d;

  // gfx1250 global_prefetch_b8: pull flow2 stream ~16KB ahead (one lane/wave)
  if ((threadIdx.x & 31u) == 0u) {
    const float* f2f = (const float*)flow2v;
    __builtin_prefetch(f2f + base + 4096, 0, 1);
    __builtin_prefetch(f2f + base + DHW + 4096, 0, 1);
    __builtin_prefetch(f2f + base + 2 * DHW + 4096, 0, 1);
  }

  // b128 non-temporal streaming loads of the three flow2 channels
  const v4f f2d4 = __builtin_nontemporal_load(flow2v + idx);
  const v4f f2h4 = __builtin_nontemporal_load(flow2v + idx + N4);
  const v4f f2w4 = __builtin_nontemporal_load(flow2v + idx + 2 * N4);
  const float a_d[4] = {f2d4.x, f2d4.y, f2d4.z, f2d4.w};
  const float a_h[4] = {f2h4.x, f2h4.y, f2h4.z, f2h4.w};
  const float a_w[4] = {f2w4.x, f2w4.y, f2w4.z, f2w4.w};

  // per-voxel-group constants (invariant across j)
  const float cy_i = fmaf(96.f, hc, 95.5f);
  const float cz_i = fmaf(80.f, dc, 79.5f);
  const float cy_j = fmaf(192.f / 191.f, hc, -0.5f);
  const float cz_j = fmaf(160.f / 159.f, dc, -0.5f);

  float od[4], oh[4], ow[4], df[4];

#pragma unroll
  for (int j = 0; j < 4; ++j) {
    const float wc = (float)(w0 + j);

    // sample flow1 at the (intentionally unnormalized) grid2 coords
    const float ix = fmaf(iAx, a_w[j], fmaf(80.f, wc, 79.5f));
    const float iy = fmaf(iAy, a_h[j], cy_i);
    const float iz = fmaf(iAz, a_d[j], cz_i);

    float s0, s1, s2;
    sample_flow1(flow1, ix, iy, iz, s0, s1, s2);

    od[j] = s0 + a_d[j];
    oh[j] = s1 + a_h[j];
    ow[j] = s2 + a_w[j];

    // sample src at folded normalized coords
    const float jx = fmaf(jSx, ow[j], fmaf(160.f / 159.f, wc, -0.5f));
    const float jy = fmaf(jSy, oh[j], cy_j);
    const float jz = fmaf(jSz, od[j], cz_j);

    df[j] = sample_src(src, jx, jy, jz);
  }

  // b128 non-temporal streaming stores (outputs are never re-read)
  const v4f vflow_d = {od[0], od[1], od[2], od[3]};
  const v4f vflow_h = {oh[0], oh[1], oh[2], oh[3]};
  const v4f vflow_w = {ow[0], ow[1], ow[2], ow[3]};
  const v4f vdef    = {df[0], df[1], df[2], df[3]};
  __builtin_nontemporal_store(vflow_d, out_flow + idx);
  __builtin_nontemporal_store(vflow_h, out_flow + idx + N4);
  __builtin_nontemporal_store(vflow_w, out_flow + idx + 2 * N4);
  __builtin_nontemporal_store(vdef,    out_deform + idx);
}

} // namespace

extern "C" void kernel_launch(void* const* d_in, const int* in_sizes, int n_in,
                              void* d_out, int out_size, void* d_ws, size_t ws_size,
                              hipStream_t stream) {
  (void)in_sizes; (void)n_in; (void)out_size; (void)d_ws; (void)ws_size;
  const float* src        = (const float*)d_in[0];   // [1,1,D,H,W]
  const float* flow1      = (const float*)d_in[1];   // [1,3,D,H,W]
  const v4f*   flow2v     = (const v4f*)d_in[2];     // [1,3,D,H,W] as float4
  const float* range_flow = (const float*)d_in[3];   // scalar

  v4f* out_deform = (v4f*)d_out;                     // [D*H*W]
  v4f* out_flow   = (v4f*)((float*)d_out + DHW);     // [3*D*H*W]

  const int threads = 256;             // 8 wave32 waves per block
  const int blocks  = N4 / threads;    // exact: 4800
  warp3d_fused_v6<<<blocks, threads, 0, stream>>>(
      src, flow1, flow2v, range_flow, out_deform, out_flow);
}